// RockUnitPredictor_26104811225564
// MI455X (gfx1250) — compile-verified
//
#include <hip/hip_runtime.h>
#include <hip/hip_bf16.h>

typedef __attribute__((ext_vector_type(2))) float v2f;
typedef __attribute__((ext_vector_type(8))) float v8f;

#define N_NODES 100000

// ---------------------------------------------------------------------------
// Utility: zero a float buffer
// ---------------------------------------------------------------------------
__global__ void zero_kernel(float* __restrict__ p, int n) {
    int i = blockIdx.x * blockDim.x + threadIdx.x;
    if (i < n) p[i] = 0.0f;
}

// ---------------------------------------------------------------------------
// Degree count (float, so we can reuse as mean divisor) and reciprocal
// ---------------------------------------------------------------------------
__global__ void degree_kernel(const int* __restrict__ dst,
                              float* __restrict__ deg, int E) {
    int e = blockIdx.x * blockDim.x + threadIdx.x;
    if (e < E) unsafeAtomicAdd(&deg[dst[e]], 1.0f);
}

__global__ void invdeg_kernel(float* __restrict__ deg, int n) {
    int i = blockIdx.x * blockDim.x + threadIdx.x;
    if (i < n) deg[i] = 1.0f / fmaxf(deg[i], 1.0f);
}

// ---------------------------------------------------------------------------
// Scatter-add of source-node features into per-destination accumulator.
// One wave (32 lanes) per edge; lanes split the feature dimension.
// agg (100k x 128 x 4B = 51MB) fits in the 192MB L2, so the f32 atomics
// resolve in L2.
// ---------------------------------------------------------------------------
template <int F>
__global__ void scatter_kernel(const float* __restrict__ h,
                               const int* __restrict__ src,
                               const int* __restrict__ dst,
                               float* __restrict__ agg, int E) {
    int gw   = (blockIdx.x * blockDim.x + threadIdx.x) >> 5;  // wave id
    int lane = threadIdx.x & 31;
    if (gw >= E) return;
    int s = src[gw];
    int d = dst[gw];
    if (F == 32) {
        float v = h[(size_t)s * 32 + lane];
        unsafeAtomicAdd(&agg[(size_t)d * 32 + lane], v);
    } else {  // F == 128: float4 per lane, contiguous 512B row per wave
        const float4* row = reinterpret_cast<const float4*>(h + (size_t)s * F);
        float4 v = row[lane];
        float* o = agg + (size_t)d * F + lane * 4;
        unsafeAtomicAdd(o + 0, v.x);
        unsafeAtomicAdd(o + 1, v.y);
        unsafeAtomicAdd(o + 2, v.z);
        unsafeAtomicAdd(o + 3, v.w);
    }
}

// ---------------------------------------------------------------------------
// SAGE combine via fp32 WMMA:
//   out[i,:] = prelu( (agg[i,:]*invdeg[i]) @ Wl + b + h[i,:] @ Wr )
// One block (128 threads = 4 waves) handles 16 nodes x FOUT.
// Each wave owns TPW 16x16 output tiles.  K-loop steps by 4 using
// V_WMMA_F32_16X16X4_F32, chaining two WMMAs (mean@Wl, h@Wr) into one
// accumulator.  A matrices are staged in LDS (stride FIN+1 to avoid bank
// conflicts); B comes straight from the small weight matrices (L2-resident).
// Grid is exact (100000/16 = 6250 blocks) -> EXEC all-ones for WMMA.
// ---------------------------------------------------------------------------
template <int FIN, int FOUT, int TPW, bool PRELU>
__global__ __launch_bounds__(128) void sage_combine_wmma(
    const float* __restrict__ agg, const float* __restrict__ invdeg,
    const float* __restrict__ h,
    const float* __restrict__ Wl, const float* __restrict__ bias,
    const float* __restrict__ Wr, const float* __restrict__ alpha_p,
    float* __restrict__ out) {
    constexpr int LDSTR = FIN + 1;
    __shared__ float s_mean[16 * LDSTR];
    __shared__ float s_h[16 * LDSTR];

    const int tid   = threadIdx.x;
    const int node0 = blockIdx.x * 16;

    for (int idx = tid; idx < 16 * FIN; idx += 128) {
        int r = idx / FIN;
        int c = idx % FIN;
        size_t g = (size_t)(node0 + r) * FIN + c;
        s_mean[r * LDSTR + c] = agg[g] * invdeg[node0 + r];
        s_h[r * LDSTR + c]    = h[g];
    }
    __syncthreads();

    const int wave = tid >> 5;
    const int lane = tid & 31;
    const int half = lane >> 4;   // 0: K pair {k,k+1}, 1: {k+2,k+3}
    const int col  = lane & 15;   // A: row M;  B/C/D: column N
    const float alpha = alpha_p[0];

    v8f acc[TPW];
#pragma unroll
    for (int t = 0; t < TPW; ++t)
#pragma unroll
        for (int j = 0; j < 8; ++j) acc[t][j] = 0.0f;

    const float* smrow = s_mean + col * LDSTR;  // A row for this lane
    const float* shrow = s_h    + col * LDSTR;

#pragma unroll 4
    for (int k0 = 0; k0 < FIN; k0 += 4) {
        const int kb = k0 + half * 2;
        v2f am; am.x = smrow[kb]; am.y = smrow[kb + 1];
        v2f ah; ah.x = shrow[kb]; ah.y = shrow[kb + 1];
#pragma unroll
        for (int t = 0; t < TPW; ++t) {
            const int n0 = (wave * TPW + t) * 16;
            v2f bl;
            bl.x = Wl[(size_t)kb * FOUT + n0 + col];
            bl.y = Wl[(size_t)(kb + 1) * FOUT + n0 + col];
            v2f br;
            br.x = Wr[(size_t)kb * FOUT + n0 + col];
            br.y = Wr[(size_t)(kb + 1) * FOUT + n0 + col];
            acc[t] = __builtin_amdgcn_wmma_f32_16x16x4_f32(
                false, am, false, bl, (short)0, acc[t], false, false);
            acc[t] = __builtin_amdgcn_wmma_f32_16x16x4_f32(
                false, ah, false, br, (short)0, acc[t], false, false);
        }
    }

    // Epilogue: bias + PReLU + store.  C/D layout: VGPR v, lanes 0-15 ->
    // (M=v, N=lane), lanes 16-31 -> (M=v+8, N=lane-16).
#pragma unroll
    for (int t = 0; t < TPW; ++t) {
        const int n0 = (wave * TPW + t) * 16;
        const float bv = bias[n0 + col];
#pragma unroll
        for (int v = 0; v < 8; ++v) {
            float val = acc[t][v] + bv;
            if (PRELU) val = (val >= 0.0f) ? val : alpha * val;
            const int row = node0 + v + half * 8;
            out[(size_t)row * FOUT + n0 + col] = val;
        }
    }
}

// ---------------------------------------------------------------------------
// Final classifier: out[i,:] = h3[i,:] @ Wc[64,4] + bc   (trivial FLOPs)
// ---------------------------------------------------------------------------
__global__ void classifier_kernel(const float* __restrict__ h3,
                                  const float* __restrict__ Wc,
                                  const float* __restrict__ bc,
                                  float* __restrict__ out, int n) {
    int i = blockIdx.x * blockDim.x + threadIdx.x;
    if (i >= n) return;
    float s0 = bc[0], s1 = bc[1], s2 = bc[2], s3 = bc[3];
    const float4* row = reinterpret_cast<const float4*>(h3 + (size_t)i * 64);
#pragma unroll
    for (int k4 = 0; k4 < 16; ++k4) {
        float4 v = row[k4];
        const float* w = Wc + (k4 * 4) * 4;
        s0 += v.x * w[0];  s1 += v.x * w[1];  s2 += v.x * w[2];  s3 += v.x * w[3];
        s0 += v.y * w[4];  s1 += v.y * w[5];  s2 += v.y * w[6];  s3 += v.y * w[7];
        s0 += v.z * w[8];  s1 += v.z * w[9];  s2 += v.z * w[10]; s3 += v.z * w[11];
        s0 += v.w * w[12]; s1 += v.w * w[13]; s2 += v.w * w[14]; s3 += v.w * w[15];
    }
    float* o = out + (size_t)i * 4;
    o[0] = s0; o[1] = s1; o[2] = s2; o[3] = s3;
}

// ---------------------------------------------------------------------------
extern "C" void kernel_launch(void* const* d_in, const int* in_sizes, int n_in,
                              void* d_out, int out_size, void* d_ws, size_t ws_size,
                              hipStream_t stream) {
    const float* x   = (const float*)d_in[0];
    const int*   ei  = (const int*)d_in[1];
    const float* W1l = (const float*)d_in[2];
    const float* b1  = (const float*)d_in[3];
    const float* W1r = (const float*)d_in[4];
    const float* W2l = (const float*)d_in[5];
    const float* b2  = (const float*)d_in[6];
    const float* W2r = (const float*)d_in[7];
    const float* W3l = (const float*)d_in[8];
    const float* b3  = (const float*)d_in[9];
    const float* W3r = (const float*)d_in[10];
    const float* Wc  = (const float*)d_in[11];
    const float* bc  = (const float*)d_in[12];
    const float* a   = (const float*)d_in[13];
    float* out = (float*)d_out;

    const int E = in_sizes[1] / 2;
    const int* srcv = ei;
    const int* dstv = ei + E;

    // Workspace carve-out (256B aligned): deg + agg + 2 activation buffers.
    char* ws = (char*)d_ws;
    size_t off = 0;
    auto carve = [&](size_t bytes) {
        float* p = (float*)(ws + off);
        off += (bytes + 255) & ~(size_t)255;
        return p;
    };
    float* deg = carve((size_t)N_NODES * 4);            // degree -> 1/deg
    float* agg = carve((size_t)N_NODES * 128 * 4);      // scatter target
    float* hA  = carve((size_t)N_NODES * 128 * 4);      // h1, later h3
    float* hB  = carve((size_t)N_NODES * 128 * 4);      // h2
    float* h3  = hA;                                    // h1 dead after L2

    const int TB = 256;
    const int nBlk16 = N_NODES / 16;                    // 6250, exact

    // Degrees (shared by all layers)
    zero_kernel<<<(N_NODES + TB - 1) / TB, TB, 0, stream>>>(deg, N_NODES);
    degree_kernel<<<(E + TB - 1) / TB, TB, 0, stream>>>(dstv, deg, E);
    invdeg_kernel<<<(N_NODES + TB - 1) / TB, TB, 0, stream>>>(deg, N_NODES);

    // ---- Layer 1: 32 -> 128, PReLU ----
    zero_kernel<<<(N_NODES * 32 + TB - 1) / TB, TB, 0, stream>>>(agg, N_NODES * 32);
    scatter_kernel<32><<<((size_t)E * 32 + TB - 1) / TB, TB, 0, stream>>>(
        x, srcv, dstv, agg, E);
    sage_combine_wmma<32, 128, 2, true><<<nBlk16, 128, 0, stream>>>(
        agg, deg, x, W1l, b1, W1r, a, hA);

    // ---- Layer 2: 128 -> 128, PReLU ----
    zero_kernel<<<(N_NODES * 128 + TB - 1) / TB, TB, 0, stream>>>(agg, N_NODES * 128);
    scatter_kernel<128><<<((size_t)E * 32 + TB - 1) / TB, TB, 0, stream>>>(
        hA, srcv, dstv, agg, E);
    sage_combine_wmma<128, 128, 2, true><<<nBlk16, 128, 0, stream>>>(
        agg, deg, hA, W2l, b2, W2r, a, hB);

    // ---- Layer 3: 128 -> 64, no activation ----
    zero_kernel<<<(N_NODES * 128 + TB - 1) / TB, TB, 0, stream>>>(agg, N_NODES * 128);
    scatter_kernel<128><<<((size_t)E * 32 + TB - 1) / TB, TB, 0, stream>>>(
        hB, srcv, dstv, agg, E);
    sage_combine_wmma<128, 64, 1, false><<<nBlk16, 128, 0, stream>>>(
        agg, deg, hB, W3l, b3, W3r, a, h3);

    // ---- Classifier: 64 -> 4 ----
    classifier_kernel<<<(N_NODES + TB - 1) / TB, TB, 0, stream>>>(
        h3, Wc, bc, out, N_NODES);
}